// NearestMemoryManager_64501818851612
// MI455X (gfx1250) — compile-verified
//
#include <hip/hip_runtime.h>
#include <math.h>

typedef float v2f __attribute__((ext_vector_type(2)));
typedef float v8f __attribute__((ext_vector_type(8)));

static constexpr int kB    = 8;
static constexpr int kNPOS = 512;
static constexpr int kNNEG = 512;
static constexpr int kD    = 128;
static constexpr int kNLEM = 32768;
static constexpr int kSeq  = kNPOS + kNNEG;   // 1024 rows of x per batch

// ---------------------------------------------------------------------------
// GEMM: C[r, n] = sum_d A[r, d] * Mem[n, d]
//   A row r -> x + (b*1024 + row_off + p)*128,  b = r/512, p = r%512
//   One wave per 16-row M-tile, nt_per_wave consecutive 16-col N-tiles.
//   Core op: V_WMMA_F32_16X16X4_F32, 32 K-steps per tile in two interleaved 16-step chains.
// ---------------------------------------------------------------------------
__global__ __launch_bounds__(256) void gemm_f32_wmma(
    const float* __restrict__ X, const float* __restrict__ Mem,
    float* __restrict__ C, int row_off, int ldc, int nt_per_wave)
{
    const int lane = threadIdx.x & 31;
    const int wave = threadIdx.x >> 5;
    const int half = lane >> 4;      // 0: lanes 0-15, 1: lanes 16-31
    const int lm   = lane & 15;

    const int mtile = blockIdx.x;          // 256 M-tiles (4096 rows / 16)
    const int r0 = mtile * 16;
    const int b  = r0 / kNPOS;             // uniform per tile (512 % 16 == 0)
    const int p0 = r0 % kNPOS;

    // A fragment layout (32-bit A 16x4): lane=(k>=2)*16+m, vgpr=k&1
    // -> lane loads float2 at a_row + 4*kstep + 2*half
    const float* arow = X + ((size_t)(b * kSeq + row_off + p0 + lm)) * kD + 2 * half;

    v2f afrag[32];
#pragma unroll
    for (int k = 0; k < 32; ++k)
        afrag[k] = *(const v2f*)(arow + 4 * k);

    const int n_base = (blockIdx.y * 8 + wave) * (nt_per_wave * 16);

    for (int t = 0; t < nt_per_wave; ++t) {
        const int n0 = n_base + t * 16;

        // B fragment (4x16): lane=(k>=2)*16+n, vgpr=k&1 -> same float2 pattern
        const float* brow = Mem + (size_t)(n0 + lm) * kD + 2 * half;
        v2f bfrag[32];
#pragma unroll
        for (int k = 0; k < 32; ++k)
            bfrag[k] = *(const v2f*)(brow + 4 * k);

        v8f acc0 = {};
        v8f acc1 = {};
#pragma unroll
        for (int k = 0; k < 32; k += 2) {
            acc0 = __builtin_amdgcn_wmma_f32_16x16x4_f32(
                false, afrag[k],     false, bfrag[k],     (short)0, acc0, false, false);
            acc1 = __builtin_amdgcn_wmma_f32_16x16x4_f32(
                false, afrag[k + 1], false, bfrag[k + 1], (short)0, acc1, false, false);
        }
        v8f acc = acc0 + acc1;

        // D layout: VGPR j -> row (j + 8*half), col lm
        float* cptr = C + (size_t)(r0 + 8 * half) * ldc + (n0 + lm);
#pragma unroll
        for (int j = 0; j < 8; ++j)
            cptr[(size_t)j * ldc] = acc[j];
    }
}

// ---------------------------------------------------------------------------
// new_mem: one wave32 per memory row, float4 per lane (128 floats/row)
// ---------------------------------------------------------------------------
__global__ __launch_bounds__(256) void newmem_kernel(
    const float* __restrict__ X, const int* __restrict__ visible,
    const float* __restrict__ Mem, const int* __restrict__ lru_p,
    float* __restrict__ out)
{
    const int lane = threadIdx.x & 31;
    const int wave = threadIdx.x >> 5;
    const int row  = blockIdx.x * 8 + wave;       // 0..32767
    const int d0   = lane * 4;                    // float4 per lane

    const int lru   = *lru_p;
    const int start = kNPOS + lru * kNNEG * kB;   // 12800 for lru=3

    float4 v;
    if (row < kNPOS) {
        // pos_new = 0.5*memory + 0.5*mean_b(t * vis)
        float4 m4 = *(const float4*)(Mem + (size_t)row * kD + d0);
        float4 s = make_float4(0.f, 0.f, 0.f, 0.f);
#pragma unroll
        for (int bb = 0; bb < kB; ++bb) {
            float vis = (float)visible[bb * kNPOS + row];
            float4 t4 = *(const float4*)(X + ((size_t)bb * kSeq + row) * kD + d0);
            s.x += t4.x * vis; s.y += t4.y * vis;
            s.z += t4.z * vis; s.w += t4.w * vis;
        }
        const float inv_b = 1.0f / (float)kB;
        v.x = 0.5f * (m4.x + s.x * inv_b);
        v.y = 0.5f * (m4.y + s.y * inv_b);
        v.z = 0.5f * (m4.z + s.z * inv_b);
        v.w = 0.5f * (m4.w + s.w * inv_b);
    } else if (row >= start && row < start + kNNEG * kB) {
        int idx = row - start;
        int bb  = idx / kNNEG;
        int n   = idx % kNNEG;
        v = *(const float4*)(X + ((size_t)bb * kSeq + kNPOS + n) * kD + d0);
    } else {
        v = *(const float4*)(Mem + (size_t)row * kD + d0);
    }

    // wave32 reduction of sum of squares
    float ss = v.x * v.x + v.y * v.y + v.z * v.z + v.w * v.w;
#pragma unroll
    for (int off = 16; off > 0; off >>= 1)
        ss += __shfl_xor(ss, off, 32);

    float norm  = fmaxf(sqrtf(ss), 1e-12f);
    float scale = 1.0f / norm;
    float4 o = make_float4(v.x * scale, v.y * scale, v.z * scale, v.w * scale);
    *(float4*)(out + (size_t)row * kD + d0) = o;
}

// ---------------------------------------------------------------------------
// y_idx passthrough (int -> float slot in concatenated output)
// ---------------------------------------------------------------------------
__global__ void ycopy_kernel(const int* __restrict__ y, float* __restrict__ out)
{
    int i = blockIdx.x * blockDim.x + threadIdx.x;
    if (i < kB * kNPOS) out[i] = (float)y[i];
}

extern "C" void kernel_launch(void* const* d_in, const int* in_sizes, int n_in,
                              void* d_out, int out_size, void* d_ws, size_t ws_size,
                              hipStream_t stream)
{
    const float* x       = (const float*)d_in[0];
    const int*   y       = (const int*)  d_in[1];
    const int*   visible = (const int*)  d_in[2];
    const float* memory  = (const float*)d_in[3];
    const int*   lru     = (const int*)  d_in[4];
    float* out = (float*)d_out;

    const size_t SIM_OFF   = 0;
    const size_t Y_OFF     = SIM_OFF   + (size_t)kB * kNPOS * kNLEM;  // 134217728
    const size_t NOISE_OFF = Y_OFF     + (size_t)kB * kNPOS;          // +4096
    const size_t MEM_OFF   = NOISE_OFF + (size_t)kB * kNNEG * kNPOS;  // +2097152

    // similarity: M=4096, N=32768. 8 waves/block, 8 N-tiles/wave -> 1024 cols/block
    gemm_f32_wmma<<<dim3(256, 32), 256, 0, stream>>>(
        x, memory, out + SIM_OFF, /*row_off=*/0, /*ldc=*/kNLEM, /*nt=*/8);

    // noise_similarity: M=4096, N=512. 2 N-tiles/wave -> 256 cols/block
    gemm_f32_wmma<<<dim3(256, 2), 256, 0, stream>>>(
        x, memory, out + NOISE_OFF, /*row_off=*/kNPOS, /*ldc=*/kNPOS, /*nt=*/2);

    ycopy_kernel<<<16, 256, 0, stream>>>(y, out + Y_OFF);

    newmem_kernel<<<4096, 256, 0, stream>>>(x, visible, memory, lru, out + MEM_OFF);
}